// StructuredGaussianMarkovPosterior_25503515804374
// MI455X (gfx1250) — compile-verified
//
#include <hip/hip_runtime.h>
#include <hip/hip_bf16.h>

// Problem constants (from reference): N=8192, T=512, S=4, ND=8, NC=16
#define N_ 8192
#define T_ 512
#define S_ 4
#define ND_ 8
#define NC_ 16

typedef __attribute__((ext_vector_type(2))) float v2f;
typedef __attribute__((ext_vector_type(8))) float v8f;

// ---------------------------------------------------------------------------
// Kernel 1: proj[r] = drivers[r,:8]·gamma + cov[r,:16]·alpha  for r in [0,N*T)
// via V_WMMA_F32_16X16X4_F32.  One wave handles 16 consecutive rows.
//   B[k, col] = x[row_col, K0+k]  (data)   -> lane loads float2 of its row
//   A[m, k]   = w[K0+k]           (weights broadcast over M)
//   D[m, col] = proj(row_col) for every m  -> lanes 0..15 store c[0] coalesced
// ---------------------------------------------------------------------------
__global__ __launch_bounds__(256) void proj_wmma_kernel(
    const float* __restrict__ drivers,   // [N*T, 8]
    const float* __restrict__ cov,       // [N*T, 16]
    const float* __restrict__ gamma_w,   // [8]
    const float* __restrict__ alpha_w,   // [16]
    float* __restrict__ proj)            // [N*T]
{
    const int  tid  = blockIdx.x * blockDim.x + threadIdx.x;
    const int  lane = tid & 31;
    const int  half = lane >> 4;       // lanes 0-15 hold K slots {0,1}; 16-31 hold {2,3}
    const int  mrow = lane & 15;       // which of the 16 rows this lane's B column is
    const long wave = (long)(tid >> 5);
    const long r0   = wave * 16;
    const long row  = r0 + mrow;

    const float* __restrict__ drow = drivers + row * ND_;
    const float* __restrict__ crow = cov     + row * NC_;

    v8f c = {0.f, 0.f, 0.f, 0.f, 0.f, 0.f, 0.f, 0.f};
    v2f a, b;
    int off;

    // driver chunk 0: features 0..3
    off = 0 + 2 * half;
    b = *reinterpret_cast<const v2f*>(drow + off);
    a = *reinterpret_cast<const v2f*>(gamma_w + off);
    c = __builtin_amdgcn_wmma_f32_16x16x4_f32(false, a, false, b, (short)0, c, false, false);

    // driver chunk 1: features 4..7
    off = 4 + 2 * half;
    b = *reinterpret_cast<const v2f*>(drow + off);
    a = *reinterpret_cast<const v2f*>(gamma_w + off);
    c = __builtin_amdgcn_wmma_f32_16x16x4_f32(false, a, false, b, (short)0, c, false, false);

    // covariate chunks: features 0..15 of cov
#pragma unroll
    for (int j = 0; j < 4; ++j) {
        off = 4 * j + 2 * half;
        b = *reinterpret_cast<const v2f*>(crow + off);
        a = *reinterpret_cast<const v2f*>(alpha_w + off);
        c = __builtin_amdgcn_wmma_f32_16x16x4_f32(false, a, false, b, (short)0, c, false, false);
    }

    // Every row m of D holds the same value for column `mrow`; lane<16 has
    // D[0..7, lane] in c[0..7].  One coalesced 64B store per wave.
    if (lane < 16) {
        proj[r0 + lane] = c[0];
    }
}

// ---------------------------------------------------------------------------
// Kernel 2: per-(s,n) sequential scan over t, float4-chunked streaming.
//   mu_t = psi*Z_{t-1} + proj[n,t] + aY*Y[n,t-1];  Z_t = mu_t + sigma*eps
// ---------------------------------------------------------------------------
__global__ __launch_bounds__(256) void scan_kernel(
    const float* __restrict__ proj,      // [N*T]
    const float* __restrict__ Y,         // [N*T]
    const float* __restrict__ eps,       // [S*N*T]
    const float* __restrict__ p_psi,
    const float* __restrict__ p_aY,
    const float* __restrict__ p_lts,
    const float* __restrict__ p_mu0,
    const float* __restrict__ p_lsi,
    float* __restrict__ Z_out,           // [S*N*T]
    float* __restrict__ mu_out,          // [S*N*T]
    float* __restrict__ lv_out)          // [S*N*T]
{
    const int g = blockIdx.x * blockDim.x + threadIdx.x;   // 0..S*N-1 = (s*N + n)
    const int n = g & (N_ - 1);

    const float psi    = p_psi[0];
    const float aY     = p_aY[0];
    const float lts    = p_lts[0];
    const float lsi    = p_lsi[0];
    const float sigma  = __expf(lts);
    const float sigma0 = __expf(lsi);
    const float muini  = p_mu0[0];
    const float lv0    = 2.0f * lsi;
    const float lvt    = 2.0f * lts;

    const size_t base_n  = (size_t)n * T_;
    const size_t base_sn = (size_t)g * T_;

    const float4* __restrict__ eps4  = reinterpret_cast<const float4*>(eps  + base_sn);
    const float4* __restrict__ proj4 = reinterpret_cast<const float4*>(proj + base_n);
    const float4* __restrict__ Y4    = reinterpret_cast<const float4*>(Y    + base_n);
    float4* __restrict__ Z4 = reinterpret_cast<float4*>(Z_out  + base_sn);
    float4* __restrict__ M4 = reinterpret_cast<float4*>(mu_out + base_sn);
    float4* __restrict__ L4 = reinterpret_cast<float4*>(lv_out + base_sn);

    float4 e4 = eps4[0];
    float4 p4 = proj4[0];
    float4 y4 = Y4[0];
    float4 z, m, l;

    // t = 0 (no lagged Y, init params)
    float mu = muini + p4.x;
    float Z  = mu + sigma0 * e4.x;
    z.x = Z; m.x = mu; l.x = lv0;
    // t = 1..3
    mu = psi * Z + p4.y + aY * y4.x; Z = mu + sigma * e4.y; z.y = Z; m.y = mu; l.y = lvt;
    mu = psi * Z + p4.z + aY * y4.y; Z = mu + sigma * e4.z; z.z = Z; m.z = mu; l.z = lvt;
    mu = psi * Z + p4.w + aY * y4.z; Z = mu + sigma * e4.w; z.w = Z; m.w = mu; l.w = lvt;
    float Yprev = y4.w;
    Z4[0] = z; M4[0] = m; L4[0] = l;

    l = make_float4(lvt, lvt, lvt, lvt);

    for (int c = 1; c < T_ / 4; ++c) {
        e4 = eps4[c]; p4 = proj4[c]; y4 = Y4[c];
        if (c + 4 < T_ / 4) {
            __builtin_prefetch((const void*)(eps4  + c + 4), 0, 0);
            __builtin_prefetch((const void*)(proj4 + c + 4), 0, 0);
            __builtin_prefetch((const void*)(Y4    + c + 4), 0, 0);
        }
        mu = psi * Z + p4.x + aY * Yprev; Z = mu + sigma * e4.x; z.x = Z; m.x = mu;
        mu = psi * Z + p4.y + aY * y4.x;  Z = mu + sigma * e4.y; z.y = Z; m.y = mu;
        mu = psi * Z + p4.z + aY * y4.y;  Z = mu + sigma * e4.z; z.z = Z; m.z = mu;
        mu = psi * Z + p4.w + aY * y4.z;  Z = mu + sigma * e4.w; z.w = Z; m.w = mu;
        Yprev = y4.w;
        Z4[c] = z; M4[c] = m; L4[c] = l;
    }
}

// ---------------------------------------------------------------------------
// Fallback (only if workspace is too small): scan with proj computed inline.
// ---------------------------------------------------------------------------
__global__ __launch_bounds__(256) void scan_fallback_kernel(
    const float* __restrict__ drivers, const float* __restrict__ cov,
    const float* __restrict__ Y, const float* __restrict__ eps,
    const float* __restrict__ p_psi, const float* __restrict__ gamma_w,
    const float* __restrict__ alpha_w, const float* __restrict__ p_aY,
    const float* __restrict__ p_lts, const float* __restrict__ p_mu0,
    const float* __restrict__ p_lsi,
    float* __restrict__ Z_out, float* __restrict__ mu_out, float* __restrict__ lv_out)
{
    const int g = blockIdx.x * blockDim.x + threadIdx.x;
    const int n = g & (N_ - 1);

    const float psi    = p_psi[0];
    const float aY     = p_aY[0];
    const float lts    = p_lts[0];
    const float lsi    = p_lsi[0];
    const float sigma  = __expf(lts);
    const float sigma0 = __expf(lsi);
    const float muini  = p_mu0[0];
    const float lv0    = 2.0f * lsi;
    const float lvt    = 2.0f * lts;

    float gw[ND_], aw[NC_];
#pragma unroll
    for (int d = 0; d < ND_; ++d) gw[d] = gamma_w[d];
#pragma unroll
    for (int cc = 0; cc < NC_; ++cc) aw[cc] = alpha_w[cc];

    const size_t base_n  = (size_t)n * T_;
    const size_t base_sn = (size_t)g * T_;

    float Z = 0.f, Yprev = 0.f;
    for (int t = 0; t < T_; ++t) {
        const float* dp = drivers + (base_n + t) * ND_;
        const float* cp = cov     + (base_n + t) * NC_;
        float p = 0.f;
#pragma unroll
        for (int d = 0; d < ND_; ++d) p += dp[d] * gw[d];
#pragma unroll
        for (int cc = 0; cc < NC_; ++cc) p += cp[cc] * aw[cc];

        float mu, lv;
        if (t == 0) {
            mu = muini + p;
            Z  = mu + sigma0 * eps[base_sn];
            lv = lv0;
        } else {
            mu = psi * Z + p + aY * Yprev;
            Z  = mu + sigma * eps[base_sn + t];
            lv = lvt;
        }
        Yprev = Y[base_n + t];
        Z_out [base_sn + t] = Z;
        mu_out[base_sn + t] = mu;
        lv_out[base_sn + t] = lv;
    }
}

extern "C" void kernel_launch(void* const* d_in, const int* in_sizes, int n_in,
                              void* d_out, int out_size, void* d_ws, size_t ws_size,
                              hipStream_t stream) {
    const float* drivers        = (const float*)d_in[0];
    const float* covariates     = (const float*)d_in[1];
    const float* Y              = (const float*)d_in[2];
    const float* eps            = (const float*)d_in[3];
    const float* tilde_psi      = (const float*)d_in[4];
    const float* gamma_w        = (const float*)d_in[5];
    const float* alpha_w        = (const float*)d_in[6];
    const float* alpha_Y_lag    = (const float*)d_in[7];
    const float* log_tilde_sig  = (const float*)d_in[8];
    const float* mu_init        = (const float*)d_in[9];
    const float* log_sigma_init = (const float*)d_in[10];

    const size_t SNT = (size_t)S_ * N_ * T_;
    float* Z_out  = (float*)d_out;
    float* mu_out = Z_out + SNT;
    float* lv_out = mu_out + SNT;

    const size_t projBytes = (size_t)N_ * T_ * sizeof(float);
    const int scanBlocks = (S_ * N_) / 256;               // 128 blocks x 256 thr

    if (ws_size >= projBytes) {
        float* proj = (float*)d_ws;
        // N*T/16 tiles, one wave (32 threads) per tile
        const long totalThreads = (long)(N_ * (long)T_ / 16) * 32;
        const int projBlocks = (int)(totalThreads / 256);  // 32768
        proj_wmma_kernel<<<projBlocks, 256, 0, stream>>>(
            drivers, covariates, gamma_w, alpha_w, proj);
        scan_kernel<<<scanBlocks, 256, 0, stream>>>(
            proj, Y, eps, tilde_psi, alpha_Y_lag, log_tilde_sig,
            mu_init, log_sigma_init, Z_out, mu_out, lv_out);
    } else {
        scan_fallback_kernel<<<scanBlocks, 256, 0, stream>>>(
            drivers, covariates, Y, eps, tilde_psi, gamma_w, alpha_w,
            alpha_Y_lag, log_tilde_sig, mu_init, log_sigma_init,
            Z_out, mu_out, lv_out);
    }
}